// GPRGNNNet1_22694607192493
// MI455X (gfx1250) — compile-verified
//
#include <hip/hip_runtime.h>
#include <hip/hip_bf16.h>

typedef __attribute__((ext_vector_type(2))) float v2f;
typedef __attribute__((ext_vector_type(8))) float v8f;

#define N_NODES 100000
#define IN_CH   500
#define HID     64
#define OUT_CH  40
#define K_HOPS  10
#define BN_EPS  1e-5f

// ---------------------------------------------------------------------------
// Kernel 1: h = relu(BN(x @ W1^T + b1))   via V_WMMA_F32_16X16X4_F32
// One wave per 16-node tile; 4 WMMA accumulators cover all 64 hidden channels
// so each x element is loaded exactly once (A fragment reused 4x).
// ---------------------------------------------------------------------------
__global__ void __launch_bounds__(256)
k_gemm1_bn_relu(const float* __restrict__ x,  const float* __restrict__ W1,
                const float* __restrict__ b1,
                const float* __restrict__ gamma, const float* __restrict__ beta,
                const float* __restrict__ mean,  const float* __restrict__ var,
                float* __restrict__ h, int nTiles)
{
    const int wave = threadIdx.x >> 5;
    const int tile = blockIdx.x * 8 + wave;
    if (tile >= nTiles) return;              // wave-uniform: EXEC stays all-ones
    const int lane = threadIdx.x & 31;
    const int r    = lane & 15;              // row-in-tile (A) / column (B,C,D)
    const int half = lane >> 4;              // K pair select: {K0,K1} vs {K2,K3}
    const int m0   = tile * 16;

    const float* xrow = x  + (size_t)(m0 + r) * IN_CH + 2 * half;
    const float* wr0  = W1 + (size_t)( 0 + r) * IN_CH + 2 * half;
    const float* wr1  = W1 + (size_t)(16 + r) * IN_CH + 2 * half;
    const float* wr2  = W1 + (size_t)(32 + r) * IN_CH + 2 * half;
    const float* wr3  = W1 + (size_t)(48 + r) * IN_CH + 2 * half;

    v8f acc0 = {}, acc1 = {}, acc2 = {}, acc3 = {};
    for (int k0 = 0; k0 < IN_CH; k0 += 4) {
        v2f a  = *(const v2f*)(xrow + k0);
        v2f b0 = *(const v2f*)(wr0  + k0);
        v2f b1v= *(const v2f*)(wr1  + k0);
        v2f b2v= *(const v2f*)(wr2  + k0);
        v2f b3v= *(const v2f*)(wr3  + k0);
        acc0 = __builtin_amdgcn_wmma_f32_16x16x4_f32(false, a, false, b0,  (short)0, acc0, false, false);
        acc1 = __builtin_amdgcn_wmma_f32_16x16x4_f32(false, a, false, b1v, (short)0, acc1, false, false);
        acc2 = __builtin_amdgcn_wmma_f32_16x16x4_f32(false, a, false, b2v, (short)0, acc2, false, false);
        acc3 = __builtin_amdgcn_wmma_f32_16x16x4_f32(false, a, false, b3v, (short)0, acc3, false, false);
    }

    // Epilogue: bias + BN(running stats) + ReLU. Column is fixed per fragment.
    v8f accs[4] = {acc0, acc1, acc2, acc3};
    const int rbase = m0 + half * 8;
#pragma unroll
    for (int t = 0; t < 4; ++t) {
        const int col = t * 16 + r;
        const float sc = rsqrtf(var[col] + BN_EPS) * gamma[col];
        const float sh = beta[col] - (mean[col] - b1[col]) * sc;
#pragma unroll
        for (int v = 0; v < 8; ++v) {
            float hv = accs[t][v] * sc + sh;
            h[(size_t)(rbase + v) * HID + col] = hv > 0.f ? hv : 0.f;
        }
    }
}

// ---------------------------------------------------------------------------
// Kernel 2: z = h @ W2^T + b2 ; hidden = temp[0] * z     (WMMA, OUT padded->48)
// ---------------------------------------------------------------------------
__global__ void __launch_bounds__(256)
k_gemm2(const float* __restrict__ h, const float* __restrict__ W2,
        const float* __restrict__ b2, const float* __restrict__ temp,
        float* __restrict__ z, float* __restrict__ hidden, int nTiles)
{
    const int wave = threadIdx.x >> 5;
    const int tile = blockIdx.x * 8 + wave;
    if (tile >= nTiles) return;
    const int lane = threadIdx.x & 31;
    const int r    = lane & 15;
    const int half = lane >> 4;
    const int m0   = tile * 16;

    // Clamp padded B rows in-bounds; their acc columns are never stored.
    const int j0 = (0  + r) < OUT_CH ? (0  + r) : OUT_CH - 1;
    const int j1 = (16 + r) < OUT_CH ? (16 + r) : OUT_CH - 1;
    const int j2 = (32 + r) < OUT_CH ? (32 + r) : OUT_CH - 1;

    const float* hrow = h  + (size_t)(m0 + r) * HID + 2 * half;
    const float* w0   = W2 + (size_t)j0 * HID + 2 * half;
    const float* w1   = W2 + (size_t)j1 * HID + 2 * half;
    const float* w2   = W2 + (size_t)j2 * HID + 2 * half;

    v8f acc0 = {}, acc1 = {}, acc2 = {};
    for (int k0 = 0; k0 < HID; k0 += 4) {
        v2f a  = *(const v2f*)(hrow + k0);
        v2f b0 = *(const v2f*)(w0 + k0);
        v2f b1v= *(const v2f*)(w1 + k0);
        v2f b2v= *(const v2f*)(w2 + k0);
        acc0 = __builtin_amdgcn_wmma_f32_16x16x4_f32(false, a, false, b0,  (short)0, acc0, false, false);
        acc1 = __builtin_amdgcn_wmma_f32_16x16x4_f32(false, a, false, b1v, (short)0, acc1, false, false);
        acc2 = __builtin_amdgcn_wmma_f32_16x16x4_f32(false, a, false, b2v, (short)0, acc2, false, false);
    }

    v8f accs[3] = {acc0, acc1, acc2};
    const float t0 = temp[0];
    const int rbase = m0 + half * 8;
#pragma unroll
    for (int t = 0; t < 3; ++t) {
        const int col = t * 16 + r;
        if (col < OUT_CH) {                    // divergence only after all WMMAs
            const float bb = b2[col];
#pragma unroll
            for (int v = 0; v < 8; ++v) {
                float zv = accs[t][v] + bb;
                size_t idx = (size_t)(rbase + v) * OUT_CH + col;
                z[idx]      = zv;
                hidden[idx] = t0 * zv;
            }
        }
    }
}

// ---------------------------------------------------------------------------
// gcn_norm: indegree (with self-loop) then in-place rsqrt
// ---------------------------------------------------------------------------
__global__ void k_deg_init(float* __restrict__ deg, int n) {
    int i = blockIdx.x * blockDim.x + threadIdx.x;
    if (i < n) deg[i] = 1.0f;                  // self-loop contribution
}

__global__ void k_deg_edges(const long long* __restrict__ ei, float* __restrict__ deg, int nE) {
    int e = blockIdx.x * blockDim.x + threadIdx.x;
    if (e < nE) {
        long long d = ei[(size_t)nE + e];      // edge_index[1][e]
        atomicAdd(&deg[d], 1.0f);
    }
}

__global__ void k_dinv(float* __restrict__ deg, int n) {
    int i = blockIdx.x * blockDim.x + threadIdx.x;
    if (i < n) { float d = deg[i]; deg[i] = d > 0.f ? rsqrtf(d) : 0.f; }
}

// ---------------------------------------------------------------------------
// Propagation hop: nxt = D^-1/2 (A+I) D^-1/2 @ cur   (COO scatter, fp32 atomics)
// ---------------------------------------------------------------------------
__global__ void k_prop_init(const float* __restrict__ dinv, const float* __restrict__ cur,
                            float* __restrict__ nxt, int nQuads) {   // nQuads = N*10
    int i = blockIdx.x * blockDim.x + threadIdx.x;
    if (i < nQuads) {
        int node = i / (OUT_CH / 4);
        float w = dinv[node]; w *= w;          // self-loop weight dinv^2
        float4 v = ((const float4*)cur)[i];
        float4 o; o.x = w * v.x; o.y = w * v.y; o.z = w * v.z; o.w = w * v.w;
        ((float4*)nxt)[i] = o;
    }
}

__global__ void __launch_bounds__(256)
k_prop_edges(const long long* __restrict__ ei, const float* __restrict__ dinv,
             const float* __restrict__ cur, float* __restrict__ nxt, int nE)
{
    int e = blockIdx.x * blockDim.x + threadIdx.x;
    if (e >= nE) return;
    const long long s = ei[e];
    const long long d = ei[(size_t)nE + e];
    const float w = dinv[s] * dinv[d];
    const float4* __restrict__ cs = (const float4*)(cur + (size_t)s * OUT_CH);
    float* __restrict__ nd = nxt + (size_t)d * OUT_CH;
#pragma unroll
    for (int q = 0; q < OUT_CH / 4; ++q) {
        float4 v = cs[q];
        atomicAdd(nd + q * 4 + 0, w * v.x);
        atomicAdd(nd + q * 4 + 1, w * v.y);
        atomicAdd(nd + q * 4 + 2, w * v.z);
        atomicAdd(nd + q * 4 + 3, w * v.w);
    }
}

__global__ void k_axpy(const float* __restrict__ temp, int kIdx,
                       const float* __restrict__ nxt, float* __restrict__ hidden, int nQuads) {
    int i = blockIdx.x * blockDim.x + threadIdx.x;
    if (i < nQuads) {
        const float t = temp[kIdx];
        float4 a = ((const float4*)hidden)[i];
        float4 v = ((const float4*)nxt)[i];
        a.x += t * v.x; a.y += t * v.y; a.z += t * v.z; a.w += t * v.w;
        ((float4*)hidden)[i] = a;
    }
}

// ---------------------------------------------------------------------------
// log_softmax over 40 channels per node
// ---------------------------------------------------------------------------
__global__ void k_logsoftmax(const float* __restrict__ hidden, float* __restrict__ out, int n) {
    int i = blockIdx.x * blockDim.x + threadIdx.x;
    if (i >= n) return;
    const float* row = hidden + (size_t)i * OUT_CH;
    float m = row[0];
#pragma unroll
    for (int c = 1; c < OUT_CH; ++c) m = fmaxf(m, row[c]);
    float s = 0.f;
#pragma unroll
    for (int c = 0; c < OUT_CH; ++c) s += __expf(row[c] - m);
    const float ls = __logf(s);
    float* o = out + (size_t)i * OUT_CH;
#pragma unroll
    for (int c = 0; c < OUT_CH; ++c) o[c] = row[c] - m - ls;
}

// ---------------------------------------------------------------------------
extern "C" void kernel_launch(void* const* d_in, const int* in_sizes, int n_in,
                              void* d_out, int out_size, void* d_ws, size_t ws_size,
                              hipStream_t stream) {
    const float*     x     = (const float*)d_in[0];
    const long long* ei    = (const long long*)d_in[1];   // int64 per reference
    const float*     W1    = (const float*)d_in[2];
    const float*     b1    = (const float*)d_in[3];
    const float*     W2    = (const float*)d_in[4];
    const float*     b2    = (const float*)d_in[5];
    const float*     gamma = (const float*)d_in[6];
    const float*     beta  = (const float*)d_in[7];
    const float*     mean  = (const float*)d_in[8];
    const float*     var   = (const float*)d_in[9];
    const float*     temp  = (const float*)d_in[10];

    const int N  = N_NODES;
    const int nE = in_sizes[1] / 2;

    // Workspace carve-out (256B aligned): h(25.6MB) curA/curB(16MB ea) hidden(16MB) deg(0.4MB)
    char* w = (char*)d_ws;
    auto carve = [&](size_t bytes) { char* p = w; w += (bytes + 255) & ~(size_t)255; return p; };
    float* h      = (float*)carve((size_t)N * HID    * sizeof(float));
    float* curA   = (float*)carve((size_t)N * OUT_CH * sizeof(float));
    float* curB   = (float*)carve((size_t)N * OUT_CH * sizeof(float));
    float* hidden = (float*)carve((size_t)N * OUT_CH * sizeof(float));
    float* deg    = (float*)carve((size_t)N * sizeof(float));   // becomes dinv in-place

    const int nTiles = N / 16;                 // 6250, N divisible by 16
    const int gTiles = (nTiles + 7) / 8;       // 8 waves (256 thr) per block
    const int nQuads = N * (OUT_CH / 4);

    k_gemm1_bn_relu<<<gTiles, 256, 0, stream>>>(x, W1, b1, gamma, beta, mean, var, h, nTiles);
    k_gemm2<<<gTiles, 256, 0, stream>>>(h, W2, b2, temp, curA, hidden, nTiles);

    k_deg_init <<<(N  + 255) / 256, 256, 0, stream>>>(deg, N);
    k_deg_edges<<<(nE + 255) / 256, 256, 0, stream>>>(ei, deg, nE);
    k_dinv     <<<(N  + 255) / 256, 256, 0, stream>>>(deg, N);

    float* cur = curA; float* nxt = curB;
    for (int k = 0; k < K_HOPS; ++k) {
        k_prop_init <<<(nQuads + 255) / 256, 256, 0, stream>>>(deg, cur, nxt, nQuads);
        k_prop_edges<<<(nE     + 255) / 256, 256, 0, stream>>>(ei, deg, cur, nxt, nE);
        k_axpy      <<<(nQuads + 255) / 256, 256, 0, stream>>>(temp, k + 1, nxt, hidden, nQuads);
        float* t = cur; cur = nxt; nxt = t;
    }

    k_logsoftmax<<<(N + 255) / 256, 256, 0, stream>>>(hidden, (float*)d_out, N);
}